// RGCNConv_56684978372939
// MI455X (gfx1250) — compile-verified
//
#include <hip/hip_runtime.h>

typedef float v2f __attribute__((ext_vector_type(2)));
typedef float v8f __attribute__((ext_vector_type(8)));

#define N_NODES 100000
#define D 128        // D_IN == D_OUT
#define R_REL 3
#define E_EDGES 1600000
#define ROWS_PER_BLOCK 32   // 2 row-tiles of 16 per block

// ---------------- zero the msg accumulator ----------------
__global__ void rgcn_zero_kernel(float* __restrict__ p, int n4) {
  int i = blockIdx.x * blockDim.x + threadIdx.x;
  if (i < n4) ((float4*)p)[i] = make_float4(0.f, 0.f, 0.f, 0.f);
}

// ---- scatter: msg[dst[e], :] += x[src[e], :] * vals[e] ----
// one wave (32 lanes) per edge, float4 (b128) gather per lane -> 128 floats/edge.
// x (51.2 MB) and msg (51.2 MB) are both L2-resident (192 MB L2), so the random
// gathers and the native global_atomic_add_f32 ops resolve in L2, not HBM.
__global__ void rgcn_scatter_kernel(const float* __restrict__ x,
                                    const int*   __restrict__ src,
                                    const int*   __restrict__ dst,
                                    const float* __restrict__ vals,
                                    float*       __restrict__ msg) {
  long long tid = (long long)blockIdx.x * blockDim.x + threadIdx.x;
  int e    = (int)(tid >> 5);
  int lane = (int)(tid & 31);
  if (e >= E_EDGES) return;
  int   s = src[e];
  int   d = dst[e];
  float v = vals[e];
  const float4 xi = ((const float4*)(x + (long long)s * D))[lane];
  float* o = msg + (long long)d * D + lane * 4;
  // unsafeAtomicAdd -> native global_atomic_add_f32 (no CAS loop)
  unsafeAtomicAdd(o + 0, xi.x * v);
  unsafeAtomicAdd(o + 1, xi.y * v);
  unsafeAtomicAdd(o + 2, xi.z * v);
  unsafeAtomicAdd(o + 3, xi.w * v);
}

// ---- out[row0:row0+32, :] (+)= A[row0:row0+32, :] @ W (128x128) ----
// 256 threads = 8 waves; wave w owns column tile w (16 cols) and BOTH row
// tiles (two independent accumulator chains -> doubles WMMA RAW distance).
// A stripe (16 KB) and full W (64 KB) staged in LDS with coalesced b128
// loads; inner loop is pure ds_load (+2addr fusion) feeding
// V_WMMA_F32_16X16X4_F32, K=128 -> 2x32 WMMAs per wave.
__launch_bounds__(256)
__global__ void rgcn_gemm_wmma_kernel(const float* __restrict__ A,
                                      const float* __restrict__ W,
                                      float*       __restrict__ out,
                                      int accumulate) {
  __shared__ float Atile[ROWS_PER_BLOCK * D];  // 16 KB
  __shared__ float Wlds[D * D];                // 64 KB  (80 KB total -> 4 WGs/WGP)
  const int row0 = blockIdx.x * ROWS_PER_BLOCK;

  {
    const float4* Ag = (const float4*)(A + (long long)row0 * D);
    float4*       As = (float4*)Atile;
    for (int i = threadIdx.x; i < ROWS_PER_BLOCK * D / 4; i += 256) As[i] = Ag[i];
    const float4* Wg = (const float4*)W;
    float4*       Ws = (float4*)Wlds;
    for (int i = threadIdx.x; i < D * D / 4; i += 256) Ws[i] = Wg[i];
  }
  __syncthreads();

  const int wave = threadIdx.x >> 5;
  const int lane = threadIdx.x & 31;
  const int half = lane >> 4;      // 0: lanes 0-15 (K k,k+1), 1: lanes 16-31 (K k+2,k+3)
  const int l16  = lane & 15;
  const int col0 = wave * 16;

  // C/D fragments: VGPR j holds row (rowBase + j + half*8), col (col0 + l16)
  v8f c0 = {0.f, 0.f, 0.f, 0.f, 0.f, 0.f, 0.f, 0.f};
  v8f c1 = {0.f, 0.f, 0.f, 0.f, 0.f, 0.f, 0.f, 0.f};
  if (accumulate) {
#pragma unroll
    for (int j = 0; j < 8; ++j) {
      c0[j] = out[(long long)(row0 +      j + half * 8) * D + col0 + l16];
      c1[j] = out[(long long)(row0 + 16 + j + half * 8) * D + col0 + l16];
    }
  }

#pragma unroll
  for (int k = 0; k < D; k += 4) {
    const int kb = k + half * 2;
    v2f b;                                    // rows kb,kb+1 of W, 512B apart in LDS
    b.x = Wlds[(kb + 0) * D + col0 + l16];    //  -> fuses to ds_load_2addr_b32
    b.y = Wlds[(kb + 1) * D + col0 + l16];
    v2f a0, a1;                               // consecutive K -> ds_load_b64/2addr
    a0.x = Atile[(l16     ) * D + kb + 0];
    a0.y = Atile[(l16     ) * D + kb + 1];
    a1.x = Atile[(l16 + 16) * D + kb + 0];
    a1.y = Atile[(l16 + 16) * D + kb + 1];
    c0 = __builtin_amdgcn_wmma_f32_16x16x4_f32(false, a0, false, b, (short)0, c0,
                                               false, false);
    c1 = __builtin_amdgcn_wmma_f32_16x16x4_f32(false, a1, false, b, (short)0, c1,
                                               false, false);
  }

#pragma unroll
  for (int j = 0; j < 8; ++j) {
    out[(long long)(row0 +      j + half * 8) * D + col0 + l16] = c0[j];
    out[(long long)(row0 + 16 + j + half * 8) * D + col0 + l16] = c1[j];
  }
}

extern "C" void kernel_launch(void* const* d_in, const int* in_sizes, int n_in,
                              void* d_out, int out_size, void* d_ws, size_t ws_size,
                              hipStream_t stream) {
  const float* x           = (const float*)d_in[0];   // [N, 128]
  const int*   src         = (const int*)  d_in[1];   // [R, E]
  const int*   dst         = (const int*)  d_in[2];   // [R, E]
  const float* vals        = (const float*)d_in[3];   // [R, E]
  const float* weight      = (const float*)d_in[4];   // [R, 128, 128]
  const float* loop_weight = (const float*)d_in[5];   // [128, 128]
  float*       out         = (float*)d_out;           // [N, 128]
  float*       msg         = (float*)d_ws;            // N*128 f32 = 51.2 MB scratch

  const dim3 blk(256);
  const int  gemm_grid = N_NODES / ROWS_PER_BLOCK;    // 3125, exact

  // out = x @ loop_weight  (init-write clears poisoned d_out)
  rgcn_gemm_wmma_kernel<<<gemm_grid, blk, 0, stream>>>(x, loop_weight, out, 0);

  const int n4 = N_NODES * D / 4;
  const unsigned zero_grid    = (unsigned)((n4 + 255) / 256);
  const unsigned scatter_grid = (unsigned)(((long long)E_EDGES * 32 + 255) / 256);

  for (int r = 0; r < R_REL; ++r) {
    rgcn_zero_kernel<<<zero_grid, blk, 0, stream>>>(msg, n4);
    rgcn_scatter_kernel<<<scatter_grid, blk, 0, stream>>>(
        x,
        src  + (long long)r * E_EDGES,
        dst  + (long long)r * E_EDGES,
        vals + (long long)r * E_EDGES,
        msg);
    rgcn_gemm_wmma_kernel<<<gemm_grid, blk, 0, stream>>>(
        msg, weight + (long long)r * D * D, out, 1);
  }
}